// RelativeSinusoidalPositionalEncoder_40149354283318
// MI455X (gfx1250) — compile-verified
//
#include <hip/hip_runtime.h>

// Problem constants (match reference: B=4, S=512, D=128, MAX_POS=128)
#define SEQ      512
#define DMODEL   128
#define NPOS     257                 // pe rows actually used: 0..256
#define ROW_BYTES (DMODEL * 4)       // 512 bytes per row
#define NTHREADS 256
#define NWAVES   8

// out[b,i,j,d] = pe[clamp(128 + j - i, 0, 256)][d]
//
// One block per i. Stage pe[0..256] (131,584 B) into LDS with async
// global->LDS loads, then stream every (b,j) output row straight from LDS to
// HBM with global_store_async_from_lds_b128: one wave-instruction moves one
// 512-B row (32 lanes x 16 B), no VGPR data movement, and ASYNCcnt keeps up
// to 63 row-stores in flight per wave. The b dimension reuses identical
// vector offsets and only swaps the scalar SADDR base.
__global__ __launch_bounds__(NTHREADS) void
relpos_bcast_kernel(const float* __restrict__ pe, float* __restrict__ out, int B)
{
    __shared__ float s_pe[NPOS * DMODEL];   // 131,584 bytes (<= 320 KB/WGP)

    const int i    = blockIdx.x;            // 0..SEQ-1
    const int t    = threadIdx.x;
    const int lane = t & 31;
    const int wave = t >> 5;

    // Wave-relative LDS byte offset of the staging buffer: low 32 bits of the
    // flat shared-aperture address (ISA 10.2).
    const unsigned lds_base = (unsigned)(size_t)(const void*)&s_pe[0];

    // ---------------- Stage pe rows [0,257) into LDS (async) ----------------
    // 257 rows * 512 B = 8224 chunks of 16 B.
    const unsigned nchunk = NPOS * (ROW_BYTES / 16);   // 8224
    for (unsigned c = t; c < nchunk; c += NTHREADS) {
        unsigned byte   = c * 16u;          // offset into pe (global) & into LDS
        unsigned ldsoff = lds_base + byte;
        asm volatile("global_load_async_to_lds_b128 %0, %1, %2 offset:0"
                     :
                     : "v"(ldsoff), "v"(byte), "s"((const void*)pe)
                     : "memory");
    }
    asm volatile("s_wait_asynccnt 0x0" ::: "memory");
    __syncthreads();

    // ------------- Stream output rows directly from LDS (async) -------------
    // 64 j's per wave; inner b loop reuses ldsoff/goff, varying only the
    // uniform 64-bit base in SADDR.
    const size_t batch_stride = (size_t)SEQ * SEQ * DMODEL;   // floats per b
    for (int j = wave; j < SEQ; j += NWAVES) {
        int rel = 128 + j - i;              // clamp(128 + (j - i), 0, 256)
        rel = rel < 0 ? 0 : (rel > 256 ? 256 : rel);

        unsigned ldsoff = lds_base + (unsigned)rel * ROW_BYTES + (unsigned)lane * 16u;
        unsigned goff   = ((unsigned)i * SEQ + (unsigned)j) * ROW_BYTES
                          + (unsigned)lane * 16u;    // max ~2^27, safe in i32

        const float* base = out;
        for (int b = 0; b < B; ++b) {
            asm volatile("global_store_async_from_lds_b128 %0, %1, %2 offset:0"
                         :
                         : "v"(goff), "v"(ldsoff), "s"((const void*)base)
                         : "memory");
            base += batch_stride;
        }
    }
    asm volatile("s_wait_asynccnt 0x0" ::: "memory");
}

extern "C" void kernel_launch(void* const* d_in, const int* in_sizes, int n_in,
                              void* d_out, int out_size, void* d_ws, size_t ws_size,
                              hipStream_t stream)
{
    (void)n_in; (void)d_ws; (void)ws_size; (void)out_size;

    // inputs (setup_inputs order): d_in[0] = x [B,512] int32 (shape only),
    //                              d_in[1] = pe [512,128] float32
    const float* pe  = (const float*)d_in[1];
    float*       out = (float*)d_out;
    const int B = in_sizes[0] / SEQ;        // 2048 / 512 = 4

    dim3 grid(SEQ);       // one block per i
    dim3 block(NTHREADS); // 8 waves of 32
    relpos_bcast_kernel<<<grid, block, 0, stream>>>(pe, out, B);
}